// SoMTP_395136991346
// MI455X (gfx1250) — compile-verified
//
#include <hip/hip_runtime.h>
#include <hip/hip_bf16.h>

typedef __attribute__((ext_vector_type(2))) float v2f;
typedef __attribute__((ext_vector_type(8))) float v8f;

#define SDTW_INF 1e10f

// Problem sizes (fixed by reference)
static constexpr int B_   = 64;
static constexpr int CIN  = 64;
static constexpr int CH   = 256;
static constexpr int T_   = 2048;
static constexpr int P_   = 10;

// Async global->LDS helpers (CDNA5 ASYNCcnt path). LDS byte offset = low 32
// bits of the generic pointer (ISA 10.2).
__device__ __forceinline__ void async_g2l_b128(unsigned lds_off, const void* gaddr) {
  asm volatile("global_load_async_to_lds_b128 %0, %1, off"
               :: "v"(lds_off), "v"(gaddr) : "memory");
}
__device__ __forceinline__ void async_g2l_b32(unsigned lds_off, const void* gaddr) {
  asm volatile("global_load_async_to_lds_b32 %0, %1, off"
               :: "v"(lds_off), "v"(gaddr) : "memory");
}
__device__ __forceinline__ void wait_async0() {
  asm volatile("s_wait_asynccnt 0" ::: "memory");
}

// ---------------------------------------------------------------------------
// K1: h = relu(conv_w @ x + b)  per batch [256x64]@[64x2048], WMMA f32 16x16x4
// grid(16, 64), block 512 (16 waves). Wave w owns M-tile w; x block staged in
// LDS via async global->LDS B128 DMA.
// ---------------------------------------------------------------------------
__global__ __launch_bounds__(512) void k_conv(const float* __restrict__ x,
                                              const float* __restrict__ W,
                                              const float* __restrict__ bias,
                                              float* __restrict__ h) {
  __shared__ float lx[64][132];                 // 64 K-rows x 128 T-cols (+pad)
  const int b    = blockIdx.y;
  const int tblk = blockIdx.x * 128;
  const float* xb = x + (size_t)b * CIN * T_;
  const unsigned lds_base = (unsigned)(size_t)(&lx[0][0]);
  // 64 rows x 32 float4 = 2048 vector elements, 4 per thread
  for (int vi = threadIdx.x; vi < 64 * 32; vi += 512) {
    int c  = vi >> 5;
    int tf = (vi & 31) * 4;
    async_g2l_b128(lds_base + (unsigned)((c * 132 + tf) * 4),
                   xb + (size_t)c * T_ + tblk + tf);
  }
  wait_async0();
  __syncthreads();

  const int wave = threadIdx.x >> 5;
  const int lane = threadIdx.x & 31;
  const int hi   = lane >> 4;                   // K-half select for A/B frags
  const int l16  = lane & 15;
  const int mrow = wave * 16;

  // Preload A fragments for this wave's 16x64 slice of W (row-major, ld=64)
  v2f afrag[16];
  const float* Wr = W + (size_t)(mrow + l16) * 64;
#pragma unroll
  for (int kk = 0; kk < 16; ++kk) {
    int k0 = kk * 4 + hi * 2;
    afrag[kk].x = Wr[k0];
    afrag[kk].y = Wr[k0 + 1];
  }
  float bv[8];
#pragma unroll
  for (int r = 0; r < 8; ++r) bv[r] = bias[mrow + r + hi * 8];

  for (int nt = 0; nt < 8; ++nt) {
    const int ncol = nt * 16 + l16;
    v8f acc = {};
#pragma unroll
    for (int kk = 0; kk < 16; ++kk) {
      int k0 = kk * 4 + hi * 2;
      v2f bfrag;
      bfrag.x = lx[k0][ncol];
      bfrag.y = lx[k0 + 1][ncol];
      acc = __builtin_amdgcn_wmma_f32_16x16x4_f32(false, afrag[kk], false, bfrag,
                                                  (short)0, acc, false, false);
    }
    float* hb = h + (size_t)b * CH * T_ + (size_t)mrow * T_ + tblk + nt * 16;
#pragma unroll
    for (int r = 0; r < 8; ++r) {
      int row = r + hi * 8;
      float v = acc[r] + bv[r];
      hb[(size_t)row * T_ + l16] = v > 0.f ? v : 0.f;
    }
  }
}

// ---------------------------------------------------------------------------
// K2: global mean (out1 base) + segment means (out2). One wave per (b,c).
// ---------------------------------------------------------------------------
__global__ __launch_bounds__(256) void k_pool(const float* __restrict__ h,
                                              float* __restrict__ out1,
                                              float* __restrict__ out2) {
  const int gwave = (blockIdx.x * 256 + threadIdx.x) >> 5;   // 0..B*CH-1
  const int lane  = threadIdx.x & 31;
  const int b = gwave >> 8, c = gwave & 255;
  const float* hp = h + ((size_t)b * CH + c) * T_;
  float seg[10];
  float tot = 0.f;
#pragma unroll
  for (int p = 0; p < P_; ++p) {
    const int start = p * 204;
    const int len   = (p == 9) ? 212 : 204;
    float s = 0.f;
    for (int t = lane; t < len; t += 32) s += hp[start + t];
    for (int off = 16; off; off >>= 1) s += __shfl_down(s, off, 32);
    seg[p] = s;
    tot += s;
  }
  if (lane == 0) {
    out1[b * CH + c] = tot * (1.f / 2048.f);
#pragma unroll
    for (int p = 0; p < P_; ++p)
      out2[(size_t)(b * CH + c) * P_ + p] = seg[p] / ((p == 9) ? 212.f : 204.f);
  }
}

// ---------------------------------------------------------------------------
// K3: D[b,p,t] = |protos[:,p]|^2 + |h[b,:,t]|^2 - 2 protos[:,p].h[b,:,t]
// grid(8, 64) block 256; protos broadcast through LDS; coalesced over t.
// ---------------------------------------------------------------------------
__global__ __launch_bounds__(256) void k_dist(const float* __restrict__ h,
                                              const float* __restrict__ protos,
                                              float* __restrict__ D) {
  __shared__ float lp[256][10];
  __shared__ float pn[10];
  const int b  = blockIdx.y;
  const int t  = blockIdx.x * 256 + threadIdx.x;
  for (int i = threadIdx.x; i < CH * P_; i += 256) lp[i / P_][i % P_] = protos[i];
  __syncthreads();
  if (threadIdx.x < P_) {
    float s = 0.f;
    for (int c = 0; c < CH; ++c) { float v = lp[c][threadIdx.x]; s += v * v; }
    pn[threadIdx.x] = s;
  }
  __syncthreads();
  const float* hp = h + (size_t)b * CH * T_ + t;
  float hn = 0.f;
  float dot[10] = {};
  for (int c = 0; c < CH; ++c) {
    float v = hp[(size_t)c * T_];
    hn = fmaf(v, v, hn);
#pragma unroll
    for (int p = 0; p < P_; ++p) dot[p] = fmaf(v, lp[c][p], dot[p]);
  }
  float* Db = D + (size_t)b * P_ * T_ + t;
#pragma unroll
  for (int p = 0; p < P_; ++p) Db[(size_t)p * T_] = pn[p] + hn - 2.f * dot[p];
}

// ---------------------------------------------------------------------------
// K4: soft-DTW forward (R) + backward gradient (A=E) + row sums.
// One wave32 per batch; lane p owns row p; anti-diagonal wavefront with shfl.
// ---------------------------------------------------------------------------
__global__ __launch_bounds__(32) void k_sdtw(const float* __restrict__ D,
                                             float* __restrict__ R,
                                             float* __restrict__ A,
                                             float* __restrict__ rowsum) {
  const int b = blockIdx.x;
  const int p = threadIdx.x;
  const bool act = p < P_;
  const float* Db = D + (size_t)b * P_ * T_;
  float* Rb = R + (size_t)b * P_ * T_;
  float* Ab = A + (size_t)b * P_ * T_;

  // forward: R[p,t] = D[p,t] + softmin(R[p-1,t], R[p,t-1], R[p-1,t-1])
  float prev1 = SDTW_INF, prev2 = SDTW_INF;     // R[p,t-1], R[p,t-2]
  for (int d = 0; d < T_ + P_ - 1; ++d) {
    const int t = d - p;
    float up     = __shfl_up(prev1, 1, 32);     // R[p-1, t]
    float upleft = __shfl_up(prev2, 1, 32);     // R[p-1, t-1]
    if (p == 0) { up = SDTW_INF; upleft = (t == 0) ? 0.f : SDTW_INF; }
    const float left = prev1;                   // R[p, t-1]
    if (act && t >= 0 && t < T_) {
      float m  = fminf(up, fminf(left, upleft));
      float sm = m - logf(expf(m - up) + expf(m - left) + expf(m - upleft));
      float v  = Db[(size_t)p * T_ + t] + sm;
      Rb[(size_t)p * T_ + t] = v;
      prev2 = prev1;
      prev1 = v;
    }
  }

  // backward: E[p,t] = sum over successors E * exp(R_s - R - D_s)
  float next1 = 0.f, next2 = 0.f;               // E[p,t+1], E[p,t+2]
  float hR1 = 0.f, hD1 = 0.f, hR2 = 0.f, hD2 = 0.f;  // (R,D) at t+1, t+2
  float rsum = 0.f;
  for (int d = T_ + P_ - 2; d >= 0; --d) {
    const int t = d - p;
    float e_up   = __shfl_down(next1, 1, 32);   // E[p+1, t]
    float e_diag = __shfl_down(next2, 1, 32);   // E[p+1, t+1]
    float Ru  = __shfl_down(hR1, 1, 32);        // R[p+1, t]
    float Du  = __shfl_down(hD1, 1, 32);
    float Rdg = __shfl_down(hR2, 1, 32);        // R[p+1, t+1]
    float Ddg = __shfl_down(hD2, 1, 32);
    if (p == P_ - 1) { e_up = 0.f; e_diag = 0.f; }
    const float e_right = next1;
    if (act && t >= 0 && t < T_) {
      const float Rown = Rb[(size_t)p * T_ + t];
      const float Down = Db[(size_t)p * T_ + t];
      float E;
      if (p == P_ - 1 && t == T_ - 1) {
        E = 1.f;
      } else {
        E = 0.f;
        if (e_up    > 0.f) E = fmaf(e_up,    expf(Ru  - Rown - Du),  E);
        if (e_right > 0.f) E = fmaf(e_right, expf(hR1 - Rown - hD1), E);
        if (e_diag  > 0.f) E = fmaf(e_diag,  expf(Rdg - Rown - Ddg), E);
      }
      Ab[(size_t)p * T_ + t] = E;
      rsum += E;
      next2 = next1; next1 = E;
      hR2 = hR1; hR1 = Rown;
      hD2 = hD1; hD1 = Down;
    }
  }
  if (act) rowsum[b * P_ + p] = rsum;
}

// ---------------------------------------------------------------------------
// K5: out3[b] = h[b] (256x2048) @ Anorm[b]^T (2048x10, padded to 16). WMMA.
// One block per batch; raw 80 KB alignment matrix DMA'd to LDS (async B32,
// coalesced global reads, transposed LDS scatter); row normalization folded
// into a per-lane reciprocal applied at B-fragment build time.
// ---------------------------------------------------------------------------
__global__ __launch_bounds__(512) void k_out3(const float* __restrict__ h,
                                              const float* __restrict__ A,
                                              const float* __restrict__ rowsum,
                                              float* __restrict__ out3) {
  extern __shared__ float lA[];                 // [T_][P_] = 81920 B
  __shared__ float rs[10];
  const int b = blockIdx.x;
  if (threadIdx.x < P_) rs[threadIdx.x] = rowsum[b * P_ + threadIdx.x];

  const float* Ab = A + (size_t)b * P_ * T_;
  const unsigned lds_base = (unsigned)(size_t)(&lA[0]);
  for (int i = threadIdx.x; i < P_ * T_; i += 512) {
    int p = i >> 11, t = i & (T_ - 1);          // coalesced reads over t
    async_g2l_b32(lds_base + (unsigned)((t * P_ + p) * 4),
                  Ab + (size_t)p * T_ + t);
  }
  wait_async0();
  __syncthreads();

  const int wave = threadIdx.x >> 5;
  const int lane = threadIdx.x & 31;
  const int hi   = lane >> 4;
  const int l16  = lane & 15;
  const int mrow = wave * 16;
  const int   pp  = (l16 < P_) ? l16 : 0;
  const float irs = (l16 < P_) ? (1.f / rs[pp]) : 0.f;   // mask folded in

  const float* hr = h + (size_t)(b * CH + mrow + l16) * T_;
  v8f acc = {};
  for (int k0 = hi * 2; k0 < T_; k0 += 4) {
    __builtin_prefetch(hr + k0 + 256, 0, 1);
    v2f a;  a.x  = hr[k0];                 a.y  = hr[k0 + 1];
    v2f bb; bb.x = lA[k0 * P_ + pp] * irs; bb.y = lA[(k0 + 1) * P_ + pp] * irs;
    acc = __builtin_amdgcn_wmma_f32_16x16x4_f32(false, a, false, bb,
                                                (short)0, acc, false, false);
  }
  if (l16 < P_) {
#pragma unroll
    for (int r = 0; r < 8; ++r) {
      int row = mrow + r + hi * 8;
      out3[(size_t)(b * CH + row) * P_ + l16] = acc[r];
    }
  }
}

// ---------------------------------------------------------------------------
// K6: attention head per batch: concat -> proj -> softmax attn -> ensemble,
// plus per-batch argmax of the 90-way switch indicator.
// ---------------------------------------------------------------------------
__global__ __launch_bounds__(256) void k_head(const float* __restrict__ out1,
                                              const float* __restrict__ out2,
                                              const float* __restrict__ out3,
                                              const float* __restrict__ sw,
                                              const float* __restrict__ enc_w,
                                              const float* __restrict__ enc_b,
                                              float* __restrict__ attn_out,
                                              float* __restrict__ ens_out,
                                              float* __restrict__ argmaxb) {
  __shared__ float co[256][32];                 // concat_out[b], padded
  __shared__ float at[30][32];
  __shared__ float ind90[90];
  const int b = blockIdx.x;
  const int c = threadIdx.x;
  const float o1 = out1[b * CH + c];
#pragma unroll
  for (int p = 0; p < P_; ++p) {
    co[c][p]      = o1;
    co[c][10 + p] = out2[(size_t)(b * CH + c) * P_ + p];
    co[c][20 + p] = out3[(size_t)(b * CH + c) * P_ + p];
  }
  __syncthreads();
  if (threadIdx.x < 30) {
    const int i = threadIdx.x;
    float s = 0.f;
    for (int cc = 0; cc < CH; ++cc) s = fmaf(co[cc][i], enc_w[cc], s);
    const float eb = enc_b[0];
    float lg[30];
    float mx = -3.4e38f;
#pragma unroll
    for (int j = 0; j < 30; ++j) { lg[j] = s * sw[j] + eb; mx = fmaxf(mx, lg[j]); }
    float ss = 0.f;
#pragma unroll
    for (int j = 0; j < 30; ++j) { lg[j] = expf(lg[j] - mx); ss += lg[j]; }
    const float inv = 1.f / ss;
    float g0 = 0.f, g1 = 0.f, g2 = 0.f;
#pragma unroll
    for (int j = 0; j < 30; ++j) {
      float a = lg[j] * inv;
      at[i][j] = a;
      attn_out[((size_t)b * 30 + i) * 30 + j] = a;
      if (j < 10) g0 += a; else if (j < 20) g1 += a; else g2 += a;
    }
    ind90[i]      = g0 * 0.1f;
    ind90[30 + i] = g1 * 0.1f;
    ind90[60 + i] = g2 * 0.1f;
  }
  __syncthreads();
  if (threadIdx.x == 0) {
    int am = 0; float bv = ind90[0];
    for (int k = 1; k < 90; ++k) if (ind90[k] > bv) { bv = ind90[k]; am = k; }
    argmaxb[b] = (float)am;
  }
  for (int j = 0; j < 30; ++j) {
    float s = 0.f;
#pragma unroll
    for (int i = 0; i < 30; ++i) s = fmaf(co[c][i], at[i][j], s);
    ens_out[(size_t)(b * CH + c) * 30 + j] = s;
  }
}

// K7: op = threshold(mean_b argmax_b)
__global__ void k_op(const float* __restrict__ argmaxb,
                     float* __restrict__ op_out, int* __restrict__ op_int) {
  if (threadIdx.x == 0) {
    float s = 0.f;
    for (int i = 0; i < B_; ++i) s += argmaxb[i];
    float ind = s * (1.f / 64.f);
    int op = (ind < 0.6f) ? 0 : ((ind < 1.6f) ? 1 : 2);
    op_out[0] = (float)op;
    op_int[0] = op;
  }
}

// K8: one = [out1|out2|out3][op]  (data-dependent branch resolved on device)
__global__ __launch_bounds__(256) void k_select(const float* __restrict__ out1,
                                                const float* __restrict__ out2,
                                                const float* __restrict__ out3,
                                                const int* __restrict__ opi,
                                                float* __restrict__ one_out) {
  const int idx = blockIdx.x * 256 + threadIdx.x;   // b*2560 + c*10 + p
  const int op = opi[0];
  const int bc = idx / P_;
  float v = (op == 0) ? out1[bc] : ((op == 1) ? out2[idx] : out3[idx]);
  one_out[idx] = v;
}

// ---------------------------------------------------------------------------
// K9/K10: fully-connected layer C[64,N] = act(Z[64,K] @ W^T + b). WMMA f32.
// W row-major [N,K]. 8 waves/block, one 16x16 tile per wave.
// ---------------------------------------------------------------------------
__global__ __launch_bounds__(256) void k_fc(const float* __restrict__ Z,
                                            const float* __restrict__ W,
                                            const float* __restrict__ bias,
                                            float* __restrict__ out,
                                            int K, int N, int relu) {
  const int wave = threadIdx.x >> 5;
  const int lane = threadIdx.x & 31;
  const int tile = blockIdx.x * 8 + wave;
  const int mt = tile & 3, nt = tile >> 2;
  const int hi = lane >> 4, l16 = lane & 15;
  const float* zr = Z + (size_t)(mt * 16 + l16) * K;
  const float* wr = W + (size_t)(nt * 16 + l16) * K;
  v8f acc = {};
  for (int k0 = hi * 2; k0 < K; k0 += 4) {
    v2f a;  a.x  = zr[k0]; a.y  = zr[k0 + 1];
    v2f bb; bb.x = wr[k0]; bb.y = wr[k0 + 1];
    acc = __builtin_amdgcn_wmma_f32_16x16x4_f32(false, a, false, bb,
                                                (short)0, acc, false, false);
  }
#pragma unroll
  for (int r = 0; r < 8; ++r) {
    int row = mt * 16 + r + hi * 8;
    int col = nt * 16 + l16;
    float v = acc[r] + bias[col];
    if (relu) v = fmaxf(v, 0.f);
    out[(size_t)row * N + col] = v;
  }
}

// K11: logits = z2 @ w3^T + b3   ([64,1024]@[1024,10], tiny)
__global__ __launch_bounds__(64) void k_fc3(const float* __restrict__ z2,
                                            const float* __restrict__ w3,
                                            const float* __restrict__ b3,
                                            float* __restrict__ out) {
  const int idx = blockIdx.x * 64 + threadIdx.x;    // 640
  if (idx >= B_ * P_) return;
  const int b = idx / P_, o = idx % P_;
  const float* z = z2 + (size_t)b * 1024;
  const float* w = w3 + (size_t)o * 1024;
  float s = b3[o];
  for (int k = 0; k < 1024; ++k) s = fmaf(z[k], w[k], s);
  out[b * P_ + o] = s;
}

// ---------------------------------------------------------------------------
extern "C" void kernel_launch(void* const* d_in, const int* in_sizes, int n_in,
                              void* d_out, int out_size, void* d_ws, size_t ws_size,
                              hipStream_t stream) {
  const float* x      = (const float*)d_in[0];
  const float* conv_w = (const float*)d_in[1];
  const float* conv_b = (const float*)d_in[2];
  const float* protos = (const float*)d_in[3];
  const float* sw     = (const float*)d_in[4];
  const float* enc_w  = (const float*)d_in[5];
  const float* enc_b  = (const float*)d_in[6];
  const float* dw1    = (const float*)d_in[7];
  const float* db1    = (const float*)d_in[8];
  const float* dw2    = (const float*)d_in[9];
  const float* db2    = (const float*)d_in[10];
  const float* dw3    = (const float*)d_in[11];
  const float* db3    = (const float*)d_in[12];

  float* out = (float*)d_out;
  // d_out layout (return order: h, out, ensemble, one, op, attn)
  float* d_h    = out;                            // 64*256*2048
  float* d_mlp  = out + 33554432;                 // 64*10
  float* d_ens  = out + 33555072;                 // 64*256*1*30
  float* d_one  = out + 34046592;                 // 64*256*10
  float* d_op   = out + 34210432;                 // 1
  float* d_attn = out + 34210433;                 // 64*30*30

  // workspace layout (floats)
  float* wsf = (float*)d_ws;
  float* w_out1 = wsf;                            // 16384
  float* w_out2 = wsf + 16384;                    // 163840
  float* w_D    = wsf + 180224;                   // 1310720
  float* w_R    = wsf + 1490944;                  // 1310720
  float* w_A    = wsf + 2801664;                  // 1310720
  float* w_rs   = wsf + 4112384;                  // 640
  float* w_out3 = wsf + 4113024;                  // 163840
  float* w_amx  = wsf + 4276864;                  // 64
  float* w_z1   = wsf + 4276928;                  // 64*512
  float* w_z2   = wsf + 4309696;                  // 64*1024
  int*   w_opi  = (int*)(wsf + 4375232);          // 1

  // 1) conv + relu (WMMA, async x staging)
  k_conv<<<dim3(16, B_), 512, 0, stream>>>(x, conv_w, conv_b, d_h);
  // 2) global/segment pooling
  k_pool<<<(B_ * CH) / 8, 256, 0, stream>>>(d_h, w_out1, w_out2);
  // 3) pairwise squared-euclidean cost D
  k_dist<<<dim3(T_ / 256, B_), 256, 0, stream>>>(d_h, protos, w_D);
  // 4) soft-DTW forward + gradient (alignment A) per batch
  k_sdtw<<<B_, 32, 0, stream>>>(w_D, w_R, w_A, w_rs);
  // 5) out3 = h @ Anorm^T (WMMA, async 80 KB LDS stage of A)
  k_out3<<<B_, 512, (size_t)(T_ * P_ * sizeof(float)), stream>>>(d_h, w_A, w_rs, w_out3);
  // 6) attention head: attn, ensemble, per-batch argmax
  k_head<<<B_, 256, 0, stream>>>(w_out1, w_out2, w_out3, sw, enc_w, enc_b,
                                 d_attn, d_ens, w_amx);
  // 7) op scalar
  k_op<<<1, 64, 0, stream>>>(w_amx, d_op, w_opi);
  // 8) branch select -> "one" (also the MLP input z)
  k_select<<<(B_ * CH * P_) / 256, 256, 0, stream>>>(w_out1, w_out2, w_out3, w_opi, d_one);
  // 9) decoder MLP (WMMA FC layers)
  k_fc<<<16, 256, 0, stream>>>(d_one, dw1, db1, w_z1, 2560, 512, 1);
  k_fc<<<32, 256, 0, stream>>>(w_z1, dw2, db2, w_z2, 512, 1024, 1);
  k_fc3<<<10, 64, 0, stream>>>(w_z2, dw3, db3, d_mlp);
}